// Model_paths_43224550867570
// MI455X (gfx1250) — compile-verified
//
#include <hip/hip_runtime.h>
#include <math.h>

#define NNODES 100000
#define INDIM  256
#define HDIM   64
#define NADJS  4
#define NNZ    1600000
#define NTYPE  4
#define NCLS   8
#define LN_EPSF 1e-5f

typedef float v2f __attribute__((ext_vector_type(2)));
typedef float v8f __attribute__((ext_vector_type(8)));

// ---------------------------------------------------------------------------
// Small helpers
// ---------------------------------------------------------------------------
__device__ __forceinline__ void softmax_small(float* v, int n) {
    float m = v[0];
    for (int i = 1; i < n; ++i) m = fmaxf(m, v[i]);
    float s = 0.f;
    for (int i = 0; i < n; ++i) { v[i] = expf(v[i] - m); s += v[i]; }
    float inv = 1.f / s;
    for (int i = 0; i < n; ++i) v[i] *= inv;
}

__device__ __forceinline__ void atomic_add_f32(float* p, float v) {
    unsafeAtomicAdd(p, v);   // lowers to global_atomic_add_f32 on gfx1250
}

// ---------------------------------------------------------------------------
// Kernel: compute all softmaxed architecture weights and fold them (together
// with the 1/|S_sub| divisors and RATIO) into a per-(pass, adjacency, buffer)
// scalar table: wtab[pass*12 + a*3 + b].
//   pass 0: S0   -> buf0=s1, buf1=s2acc(res), buf2=lr0acc
//   pass 1: S1   -> buf0=s2acc(seq), buf1=lr0acc
//   pass 2: S2   -> buf0=outseq0 (adjs 0,1 only)
//   pass 3: T0   -> buf0=t1, buf1=lr1acc
//   pass 4: T1   -> buf0=outseq1 (adjs 0,1 only)
// ---------------------------------------------------------------------------
__global__ void prep_weights_kernel(const float* __restrict__ as_seq0,
                                    const float* __restrict__ as_last_seq0,
                                    const float* __restrict__ as_res0,
                                    const float* __restrict__ as_last_res0,
                                    const float* __restrict__ as_seq1,
                                    const float* __restrict__ as_last_seq1,
                                    const float* __restrict__ as_last_res1,
                                    float* __restrict__ wtab) {
    if (threadIdx.x != 0 || blockIdx.x != 0) return;

    float wseq0a[3] = { as_seq0[0], as_seq0[1], as_seq0[2] };
    float wseq0b[3] = { as_seq0[3], as_seq0[4], as_seq0[5] };
    softmax_small(wseq0a, 3); softmax_small(wseq0b, 3);

    float wls0[2] = { as_last_seq0[0], as_last_seq0[1] };
    softmax_small(wls0, 2);

    float wres0[4] = { as_res0[0], as_res0[1], as_res0[2], as_res0[3] };
    softmax_small(wres0, 4);

    float wlr0a[3] = { as_last_res0[0], as_last_res0[1], as_last_res0[2] };
    float wlr0b[3] = { as_last_res0[3], as_last_res0[4], as_last_res0[5] };
    softmax_small(wlr0a, 3); softmax_small(wlr0b, 3);

    float wseq1[3] = { as_seq1[0], as_seq1[1], as_seq1[2] };
    softmax_small(wseq1, 3);

    float wls1[2] = { as_last_seq1[0], as_last_seq1[1] };
    softmax_small(wls1, 2);

    float wlr1[3] = { as_last_res1[0], as_last_res1[1], as_last_res1[2] };
    softmax_small(wlr1, 3);

    const float i3 = 1.f / 3.f, i4 = 0.25f, i2 = 0.5f;
    for (int a = 0; a < 4; ++a) {
        // pass 0: S0
        wtab[0 * 12 + a * 3 + 0] = (a < 3) ? wseq0a[a] * i3 : 0.f;
        wtab[0 * 12 + a * 3 + 1] = wres0[a] * i4;                       // RATIO==1
        wtab[0 * 12 + a * 3 + 2] = (a == 0) ? wlr0a[0] * i3
                                  : (a == 1) ? wlr0a[1] * i3
                                  : (a == 3) ? wlr0a[2] * i3 : 0.f;
        // pass 1: S1
        wtab[1 * 12 + a * 3 + 0] = (a < 3) ? wseq0b[a] * i3 : 0.f;
        wtab[1 * 12 + a * 3 + 1] = (a == 0) ? wlr0b[0] * i3
                                  : (a == 1) ? wlr0b[1] * i3
                                  : (a == 3) ? wlr0b[2] * i3 : 0.f;
        wtab[1 * 12 + a * 3 + 2] = 0.f;
        // pass 2: S2 (adjs 0,1)
        wtab[2 * 12 + a * 3 + 0] = (a < 2) ? wls0[a] * i2 : 0.f;
        wtab[2 * 12 + a * 3 + 1] = 0.f;
        wtab[2 * 12 + a * 3 + 2] = 0.f;
        // pass 3: T0
        wtab[3 * 12 + a * 3 + 0] = (a < 3) ? wseq1[a] * i3 : 0.f;
        wtab[3 * 12 + a * 3 + 1] = (a == 0) ? wlr1[0] * i3
                                  : (a == 1) ? wlr1[1] * i3
                                  : (a == 3) ? wlr1[2] * i3 : 0.f;
        wtab[3 * 12 + a * 3 + 2] = 0.f;
        // pass 4: T1 (adjs 0,1)
        wtab[4 * 12 + a * 3 + 0] = (a < 2) ? wls1[a] * i2 : 0.f;
        wtab[4 * 12 + a * 3 + 1] = 0.f;
        wtab[4 * 12 + a * 3 + 2] = 0.f;
    }
}

// ---------------------------------------------------------------------------
// Kernel: zero a region (float4 grid-stride)
// ---------------------------------------------------------------------------
__global__ void zero_kernel(float4* __restrict__ p, long n4) {
    long i = (long)blockIdx.x * blockDim.x + threadIdx.x;
    long stride = (long)gridDim.x * blockDim.x;
    float4 z; z.x = z.y = z.z = z.w = 0.f;
    for (; i < n4; i += stride) p[i] = z;
}

// ---------------------------------------------------------------------------
// Kernel: per-node-type projection with FP32 WMMA (16x16x4).
// hid[n,:] = feats[n,:] @ type_W[type[n]] + type_b[type[n]]
// One wave owns a 16x16 tile of hid; reuses each A fragment against the four
// per-type B fragments, then selects per row by node type.
// ---------------------------------------------------------------------------
__global__ __launch_bounds__(256)
void typeproj_wmma_kernel(const float* __restrict__ feats,
                          const int* __restrict__ types,
                          const float* __restrict__ tW,   // [T, IN, H]
                          const float* __restrict__ tb,   // [T, H]
                          float* __restrict__ hid) {      // [N, H]
    const int wave = (int)(((long)blockIdx.x * blockDim.x + threadIdx.x) >> 5);
    const int nColTiles = HDIM / 16;                // 4
    const int nTiles = (NNODES / 16) * nColTiles;   // 25000
    if (wave >= nTiles) return;
    const int lane = threadIdx.x & 31;
    const int rt = wave / nColTiles, ct = wave % nColTiles;
    const int r0 = rt * 16, c0 = ct * 16;
    const int half = lane >> 4, lr = lane & 15;

    v8f acc0 = {}, acc1 = {}, acc2 = {}, acc3 = {};
    for (int k = 0; k < INDIM; k += 4) {
        const int kb = k + 2 * half;
        v2f a;
        const float* ap = feats + (size_t)(r0 + lr) * INDIM + kb;
        a.x = ap[0]; a.y = ap[1];

        const float* wb = tW + (size_t)kb * HDIM + (c0 + lr);
        v2f b0, b1, b2, b3;
        b0.x = wb[0];                        b0.y = wb[HDIM];
        b1.x = wb[(size_t)INDIM * HDIM];     b1.y = wb[(size_t)INDIM * HDIM + HDIM];
        b2.x = wb[2 * (size_t)INDIM * HDIM]; b2.y = wb[2 * (size_t)INDIM * HDIM + HDIM];
        b3.x = wb[3 * (size_t)INDIM * HDIM]; b3.y = wb[3 * (size_t)INDIM * HDIM + HDIM];

        acc0 = __builtin_amdgcn_wmma_f32_16x16x4_f32(false, a, false, b0, (short)0, acc0, false, false);
        acc1 = __builtin_amdgcn_wmma_f32_16x16x4_f32(false, a, false, b1, (short)0, acc1, false, false);
        acc2 = __builtin_amdgcn_wmma_f32_16x16x4_f32(false, a, false, b2, (short)0, acc2, false, false);
        acc3 = __builtin_amdgcn_wmma_f32_16x16x4_f32(false, a, false, b3, (short)0, acc3, false, false);
    }

#pragma unroll
    for (int i = 0; i < 8; ++i) {
        const int node = r0 + i + 8 * half;
        const int t = types[node];
        float v = (t == 0) ? acc0[i] : (t == 1) ? acc1[i] : (t == 2) ? acc2[i] : acc3[i];
        v += tb[t * HDIM + c0 + lr];
        hid[(size_t)node * HDIM + c0 + lr] = v;
    }
}

// ---------------------------------------------------------------------------
// Kernel: C[N,64] = A[N,64] @ W[64,64] + bias   (FP32 WMMA)
// ---------------------------------------------------------------------------
__global__ __launch_bounds__(256)
void gemm64_wmma_kernel(const float* __restrict__ A,
                        const float* __restrict__ W,
                        const float* __restrict__ bias,
                        float* __restrict__ C) {
    const int wave = (int)(((long)blockIdx.x * blockDim.x + threadIdx.x) >> 5);
    const int nColTiles = HDIM / 16;
    const int nTiles = (NNODES / 16) * nColTiles;
    if (wave >= nTiles) return;
    const int lane = threadIdx.x & 31;
    const int rt = wave / nColTiles, ct = wave % nColTiles;
    const int r0 = rt * 16, c0 = ct * 16;
    const int half = lane >> 4, lr = lane & 15;

    v8f acc = {};
    for (int k = 0; k < HDIM; k += 4) {
        const int kb = k + 2 * half;
        v2f a, b;
        const float* ap = A + (size_t)(r0 + lr) * HDIM + kb;
        a.x = ap[0]; a.y = ap[1];
        const float* wp = W + (size_t)kb * HDIM + (c0 + lr);
        b.x = wp[0]; b.y = wp[HDIM];
        acc = __builtin_amdgcn_wmma_f32_16x16x4_f32(false, a, false, b, (short)0, acc, false, false);
    }
    const float bb = bias[c0 + lr];
#pragma unroll
    for (int i = 0; i < 8; ++i) {
        const int row = r0 + i + 8 * half;
        C[(size_t)row * HDIM + c0 + lr] = acc[i] + bb;
    }
}

// ---------------------------------------------------------------------------
// Kernel: fused weighted COO SpMM scatter.
// For each edge (r,c,v) of adjacency a: for each live buffer b,
//   out_b[r,:] += wtab[a*3+b] * v * x[c,:]
// One wave per edge; each lane moves two features (float2).
// Edge streams (rows/cols/vals) are read exactly once per pass -> load them
// non-temporally (TH=NT) so they do not evict the L2-resident feature
// buffers (gather source + atomic scatter targets ~100MB vs 192MB L2).
// ---------------------------------------------------------------------------
__global__ __launch_bounds__(256)
void spmm_kernel(const int* __restrict__ rows,
                 const int* __restrict__ cols,
                 const float* __restrict__ vals,
                 const float* __restrict__ x,
                 float* __restrict__ o0,
                 float* __restrict__ o1,
                 float* __restrict__ o2,
                 const float* __restrict__ wtab,
                 int nadj) {
    const long wave = ((long)blockIdx.x * blockDim.x + threadIdx.x) >> 5;
    if (wave >= (long)nadj * NNZ) return;
    const int lane = threadIdx.x & 31;
    const int a = (int)(wave / NNZ);

    const int   r = __builtin_nontemporal_load(rows + wave);
    const int   c = __builtin_nontemporal_load(cols + wave);
    const float v = __builtin_nontemporal_load(vals + wave);

    const float* xp = x + (size_t)c * HDIM + 2 * lane;
    const float x0 = xp[0] * v;
    const float x1 = xp[1] * v;

    float w;
    w = wtab[a * 3 + 0];
    if (o0 && w != 0.f) {
        float* op = o0 + (size_t)r * HDIM + 2 * lane;
        atomic_add_f32(op + 0, w * x0);
        atomic_add_f32(op + 1, w * x1);
    }
    w = wtab[a * 3 + 1];
    if (o1 && w != 0.f) {
        float* op = o1 + (size_t)r * HDIM + 2 * lane;
        atomic_add_f32(op + 0, w * x0);
        atomic_add_f32(op + 1, w * x1);
    }
    w = wtab[a * 3 + 2];
    if (o2 && w != 0.f) {
        float* op = o2 + (size_t)r * HDIM + 2 * lane;
        atomic_add_f32(op + 0, w * x0);
        atomic_add_f32(op + 1, w * x1);
    }
}

// ---------------------------------------------------------------------------
// Kernel: dst = gelu_exact(layernorm(seq + res)); one wave per node row.
// ---------------------------------------------------------------------------
__global__ __launch_bounds__(256)
void ln_gelu_kernel(const float* __restrict__ seq,
                    const float* __restrict__ res,
                    float* __restrict__ dst) {
    const long wave = ((long)blockIdx.x * blockDim.x + threadIdx.x) >> 5;
    if (wave >= NNODES) return;
    const int lane = threadIdx.x & 31;
    const size_t base = (size_t)wave * HDIM + 2 * lane;

    float v0 = seq[base + 0] + res[base + 0];
    float v1 = seq[base + 1] + res[base + 1];

    float s = v0 + v1;
#pragma unroll
    for (int m = 1; m <= 16; m <<= 1) s += __shfl_xor(s, m, 32);
    const float mean = s * (1.f / HDIM);

    const float d0 = v0 - mean, d1 = v1 - mean;
    float q = d0 * d0 + d1 * d1;
#pragma unroll
    for (int m = 1; m <= 16; m <<= 1) q += __shfl_xor(q, m, 32);
    const float inv = rsqrtf(q * (1.f / HDIM) + LN_EPSF);

    float y0 = d0 * inv, y1 = d1 * inv;
    const float kInvSqrt2 = 0.70710678118654752f;
    y0 = 0.5f * y0 * (1.f + erff(y0 * kInvSqrt2));
    y1 = 0.5f * y1 * (1.f + erff(y1 * kInvSqrt2));
    dst[base + 0] = y0;
    dst[base + 1] = y1;
}

// ---------------------------------------------------------------------------
// Kernel: attention score per node:
//   score[n] = tanh(X[n,:] @ W1 + b1) . W2 + b2
// One wave per 16-row block: 4 WMMA accumulators (one per 16-col tile) over
// K=64, then tanh + dot + cross-lane reduce.
// ---------------------------------------------------------------------------
__global__ __launch_bounds__(256)
void attn_score_kernel(const float* __restrict__ X,
                       const float* __restrict__ W1,
                       const float* __restrict__ b1,
                       const float* __restrict__ W2,   // [H,1]
                       const float* __restrict__ b2,   // [1]
                       float* __restrict__ score) {
    const int wave = (int)(((long)blockIdx.x * blockDim.x + threadIdx.x) >> 5);
    if (wave >= NNODES / 16) return;
    const int lane = threadIdx.x & 31;
    const int r0 = wave * 16;
    const int half = lane >> 4, lr = lane & 15;

    v8f acc[4] = {{}, {}, {}, {}};
    for (int k = 0; k < HDIM; k += 4) {
        const int kb = k + 2 * half;
        v2f a;
        const float* ap = X + (size_t)(r0 + lr) * HDIM + kb;
        a.x = ap[0]; a.y = ap[1];
#pragma unroll
        for (int t = 0; t < 4; ++t) {
            v2f b;
            const float* wp = W1 + (size_t)kb * HDIM + (t * 16 + lr);
            b.x = wp[0]; b.y = wp[HDIM];
            acc[t] = __builtin_amdgcn_wmma_f32_16x16x4_f32(false, a, false, b, (short)0, acc[t], false, false);
        }
    }

    float part[8];
#pragma unroll
    for (int i = 0; i < 8; ++i) {
        float s = 0.f;
#pragma unroll
        for (int t = 0; t < 4; ++t) {
            const int col = t * 16 + lr;
            s += tanhf(acc[t][i] + b1[col]) * W2[col];
        }
        part[i] = s;
    }
#pragma unroll
    for (int m = 1; m <= 8; m <<= 1) {
#pragma unroll
        for (int i = 0; i < 8; ++i) part[i] += __shfl_xor(part[i], m, 32);
    }
    if (lr == 0) {
        const float bb = b2[0];
#pragma unroll
        for (int i = 0; i < 8; ++i) score[r0 + i + 8 * half] = part[i] + bb;
    }
}

// ---------------------------------------------------------------------------
// Kernel: semantic-attention softmax blend + classifier.
//   w = softmax([s0,s1]); out = (w0*out0 + w1*out1) @ cls_W + cls_b
// One thread per node (64x8 micro-GEMM in registers). Logits are written
// once and never re-read -> non-temporal store.
// ---------------------------------------------------------------------------
__global__ __launch_bounds__(256)
void finalize_kernel(const float* __restrict__ out0,
                     const float* __restrict__ out1,
                     const float* __restrict__ sc0,
                     const float* __restrict__ sc1,
                     const float* __restrict__ clsW,   // [H, NC]
                     const float* __restrict__ clsb,   // [NC]
                     float* __restrict__ out) {        // [N, NC]
    const int n = blockIdx.x * blockDim.x + threadIdx.x;
    if (n >= NNODES) return;
    const float s0 = sc0[n], s1 = sc1[n];
    const float m = fmaxf(s0, s1);
    const float e0 = expf(s0 - m), e1 = expf(s1 - m);
    const float inv = 1.f / (e0 + e1);
    const float w0 = e0 * inv, w1 = e1 * inv;

    float acc[NCLS];
#pragma unroll
    for (int c = 0; c < NCLS; ++c) acc[c] = clsb[c];

    const float* p0 = out0 + (size_t)n * HDIM;
    const float* p1 = out1 + (size_t)n * HDIM;
    for (int h = 0; h < HDIM; ++h) {
        const float bl = w0 * p0[h] + w1 * p1[h];
        const float* wrow = clsW + h * NCLS;
#pragma unroll
        for (int c = 0; c < NCLS; ++c) acc[c] = fmaf(bl, wrow[c], acc[c]);
    }
    float* op = out + (size_t)n * NCLS;
#pragma unroll
    for (int c = 0; c < NCLS; ++c) __builtin_nontemporal_store(acc[c], op + c);
}

// ---------------------------------------------------------------------------
// Host-side orchestration
// ---------------------------------------------------------------------------
extern "C" void kernel_launch(void* const* d_in, const int* in_sizes, int n_in,
                              void* d_out, int out_size, void* d_ws, size_t ws_size,
                              hipStream_t stream) {
    const float* feats   = (const float*)d_in[0];
    const int*   types   = (const int*)  d_in[1];
    const int*   arows   = (const int*)  d_in[2];
    const int*   acols   = (const int*)  d_in[3];
    const float* avals   = (const float*)d_in[4];
    const float* tW      = (const float*)d_in[5];
    const float* tb      = (const float*)d_in[6];
    const float* W0      = (const float*)d_in[7];
    const float* b0      = (const float*)d_in[8];
    const float* W1      = (const float*)d_in[9];
    const float* b1      = (const float*)d_in[10];
    const float* as_seq0      = (const float*)d_in[11];
    const float* as_last_seq0 = (const float*)d_in[12];
    const float* as_res0      = (const float*)d_in[13];
    const float* as_last_res0 = (const float*)d_in[14];
    const float* as_seq1      = (const float*)d_in[15];
    const float* as_last_seq1 = (const float*)d_in[16];
    const float* as_last_res1 = (const float*)d_in[17];
    const float* attn1_W = (const float*)d_in[18];
    const float* attn1_b = (const float*)d_in[19];
    const float* attn2_W = (const float*)d_in[20];
    const float* attn2_b = (const float*)d_in[21];
    const float* clsW    = (const float*)d_in[22];
    const float* clsb    = (const float*)d_in[23];

    float* ws = (float*)d_ws;
    const size_t NH = (size_t)NNODES * HDIM;
    float* hid  = ws;             // [N,H]
    float* bufA = ws + 1 * NH;    // x0 / x1
    float* bufB = ws + 2 * NH;    // s1 / t1
    float* bufC = ws + 3 * NH;    // s2acc / lr1acc
    float* bufD = ws + 4 * NH;    // lr0acc
    float* bufE = ws + 5 * NH;    // outseq0 / outseq1
    float* o0b  = ws + 6 * NH;    // out0 (post LN+GELU)
    float* o1b  = ws + 7 * NH;    // out1
    float* sc0  = ws + 8 * NH;    // [N]
    float* sc1  = sc0 + NNODES;   // [N]
    float* wtab = sc1 + NNODES;   // 60 floats

    const int tileBlocks  = (NNODES / 16) * (HDIM / 16) * 32 / 256;    // 3125
    const int spmm4Blocks = (int)(((long)4 * NNZ * 32) / 256);         // 800000
    const int spmm2Blocks = (int)(((long)2 * NNZ * 32) / 256);         // 400000
    const int lnBlocks    = NNODES * 32 / 256;                         // 12500
    const int attnBlocks  = (NNODES / 16 * 32 + 255) / 256;            // 782

    // Architecture weights -> fused scalar table
    prep_weights_kernel<<<1, 32, 0, stream>>>(as_seq0, as_last_seq0, as_res0,
                                              as_last_res0, as_seq1, as_last_seq1,
                                              as_last_res1, wtab);

    // hid = per-type projection
    typeproj_wmma_kernel<<<tileBlocks, 256, 0, stream>>>(feats, types, tW, tb, hid);

    // ---- Cell 0 ----
    zero_kernel<<<2048, 256, 0, stream>>>((float4*)bufB, (long)(4 * NH / 4));   // bufB..bufE
    gemm64_wmma_kernel<<<tileBlocks, 256, 0, stream>>>(hid, W0, b0, bufA);      // x0
    // S0 pass: s1 (bufB), s2 res part (bufC), last-res S0 part (bufD)
    spmm_kernel<<<spmm4Blocks, 256, 0, stream>>>(arows, acols, avals, bufA,
                                                 bufB, bufC, bufD, wtab + 0, 4);
    // S1 pass: s2 seq part (bufC += ...), last-res S1 part (bufD += ...)
    spmm_kernel<<<spmm4Blocks, 256, 0, stream>>>(arows, acols, avals, bufB,
                                                 bufC, bufD, nullptr, wtab + 12, 4);
    // S2 pass: outseq0 (bufE), adjacencies 0,1 only
    spmm_kernel<<<spmm2Blocks, 256, 0, stream>>>(arows, acols, avals, bufC,
                                                 bufE, nullptr, nullptr, wtab + 24, 2);
    ln_gelu_kernel<<<lnBlocks, 256, 0, stream>>>(bufE, bufD, o0b);

    // ---- Cell 1 ----
    zero_kernel<<<2048, 256, 0, stream>>>((float4*)bufB, (long)(4 * NH / 4));   // bufB..bufE
    gemm64_wmma_kernel<<<tileBlocks, 256, 0, stream>>>(hid, W1, b1, bufA);      // x1
    // T0 pass: t1 (bufB), last-res T0 part (bufC)
    spmm_kernel<<<spmm4Blocks, 256, 0, stream>>>(arows, acols, avals, bufA,
                                                 bufB, bufC, nullptr, wtab + 36, 4);
    // T1 pass: outseq1 (bufE), adjacencies 0,1 only
    spmm_kernel<<<spmm2Blocks, 256, 0, stream>>>(arows, acols, avals, bufB,
                                                 bufE, nullptr, nullptr, wtab + 48, 2);
    ln_gelu_kernel<<<lnBlocks, 256, 0, stream>>>(bufE, bufC, o1b);

    // ---- Semantic attention + classifier ----
    attn_score_kernel<<<attnBlocks, 256, 0, stream>>>(o0b, attn1_W, attn1_b,
                                                      attn2_W, attn2_b, sc0);
    attn_score_kernel<<<attnBlocks, 256, 0, stream>>>(o1b, attn1_W, attn1_b,
                                                      attn2_W, attn2_b, sc1);
    finalize_kernel<<<(NNODES + 255) / 256, 256, 0, stream>>>(o0b, o1b, sc0, sc1,
                                                              clsW, clsb, (float*)d_out);
}